// GaussianRecuModel_64192581206141
// MI455X (gfx1250) — compile-verified
//
#include <hip/hip_runtime.h>

// ---------------- problem constants ----------------
static constexpr int   B_      = 512;
static constexpr int   T_      = 8192;
static constexpr int   NCHUNK  = 128;   // chunks per batch row
static constexpr int   L_      = 64;    // steps per chunk  (NCHUNK*L == T)
static constexpr int   S_      = 8;     // steps staged per round
static constexpr int   NS_     = L_ / S_; // stages per chunk (8)
static constexpr int   CPW     = 32;    // chunks per wave (== wave32 lanes)
static constexpr int   WPB     = 4;     // waves per block (128 threads)
static constexpr int   XSTRIDE = 36;    // floats per chunk row in xic LDS tile (144B, bank-padded)
static constexpr int   DSTRIDE = 18;    // floats per chunk row in dy  LDS tile (72B, bank-padded)
static constexpr float DT_     = 1e-3f;

#define AS1 __attribute__((address_space(1)))
#define AS3 __attribute__((address_space(3)))

typedef int v4i __attribute__((ext_vector_type(4)));
typedef int v2i __attribute__((ext_vector_type(2)));

// ---------------- CDNA5 async global->LDS helpers ----------------
__device__ __forceinline__ void async_g2l_b128(const float* g, float* l) {
#if __has_builtin(__builtin_amdgcn_global_load_async_to_lds_b128)
    __builtin_amdgcn_global_load_async_to_lds_b128((AS1 v4i*)g, (AS3 v4i*)l, 0, 0);
#else
    unsigned loff = (unsigned)(unsigned long long)(AS3 void*)l;
    asm volatile("global_load_async_to_lds_b128 %0, %1, off" :: "v"(loff), "v"(g) : "memory");
#endif
}

__device__ __forceinline__ void async_g2l_b64(const float* g, float* l) {
#if __has_builtin(__builtin_amdgcn_global_load_async_to_lds_b64)
    __builtin_amdgcn_global_load_async_to_lds_b64((AS1 v2i*)g, (AS3 v2i*)l, 0, 0);
#else
    unsigned loff = (unsigned)(unsigned long long)(AS3 void*)l;
    asm volatile("global_load_async_to_lds_b64 %0, %1, off" :: "v"(loff), "v"(g) : "memory");
#endif
}

__device__ __forceinline__ void wait_async0() {
#if __has_builtin(__builtin_amdgcn_s_wait_asynccnt)
    __builtin_amdgcn_s_wait_asynccnt(0);
#else
    asm volatile("s_wait_asynccnt 0" ::: "memory");
#endif
}

// Async loads complete in order: <=16 outstanding  ==>  the previous stage's 16 have landed.
__device__ __forceinline__ void wait_async16() {
#if __has_builtin(__builtin_amdgcn_s_wait_asynccnt)
    __builtin_amdgcn_s_wait_asynccnt(16);
#else
    asm volatile("s_wait_asynccnt 16" ::: "memory");
#endif
}

__device__ __forceinline__ void wait_ds0() {
    asm volatile("s_wait_dscnt 0" ::: "memory");   // LDS reads drained before async engine rewrites tile
}

// Stage S_ steps of all 32 chunks owned by this wave into LDS (16 async instrs).
// q = i*32+lane; j = q>>3 (chunk-in-wave), r = q&7 (step-in-stage):
// each 8-lane group covers one contiguous 128B line of xicovs (perfect line use).
__device__ __forceinline__ void stage_load(const float* __restrict__ xicB,
                                           const float* __restrict__ dyB,
                                           float* sx, float* sd,
                                           int lane, int stepBase) {
#pragma unroll
    for (int i = 0; i < 8; ++i) {
        int q = i * 32 + lane;
        int j = q >> 3, r = q & 7;
        int gs = stepBase + j * L_ + r;
        async_g2l_b128(xicB + (size_t)gs * 4, sx + j * XSTRIDE + r * 4);
    }
#pragma unroll
    for (int i = 0; i < 8; ++i) {
        int q = i * 32 + lane;
        int j = q >> 3, r = q & 7;
        int gs = stepBase + j * L_ + r;
        async_g2l_b64(dyB + (size_t)gs * 2, sd + j * DSTRIDE + r * 2);
    }
}

// ---------------- phase 1: per-chunk affine composition ----------------
__global__ __launch_bounds__(128) void gr_phase1(const float* __restrict__ xic,
                                                 const float* __restrict__ dy,
                                                 const float* __restrict__ A4,
                                                 const float* __restrict__ C4,
                                                 float* __restrict__ chunkMV) {
    __shared__ float sXic[WPB][2][CPW * XSTRIDE];
    __shared__ float sDy [WPB][2][CPW * DSTRIDE];

    const int wave = threadIdx.x >> 5, lane = threadIdx.x & 31;
    const int wgid = blockIdx.x * WPB + wave;          // 0..2047
    const int b    = wgid >> 2;                        // NCHUNK/CPW == 4 wave-groups per b
    const int cg   = wgid & 3;
    const int base = cg * CPW * L_;

    const float* xicB = xic + (size_t)b * T_ * 4;
    const float* dyB  = dy  + (size_t)b * T_ * 2;

    float Aeff[2][2], Cd[2][2];
#pragma unroll
    for (int i = 0; i < 2; ++i)
#pragma unroll
        for (int j = 0; j < 2; ++j) {
            Aeff[i][j] = A4[i * 2 + j] * DT_ + (i == j ? 1.0f : 0.0f);
            Cd[i][j]   = C4[i * 2 + j] * DT_;
        }

    float M00 = 1.f, M01 = 0.f, M10 = 0.f, M11 = 1.f, v0 = 0.f, v1 = 0.f;

    // prologue: fill buffer 0
    stage_load(xicB, dyB, sXic[wave][0], sDy[wave][0], lane, base);

    for (int k = 0; k < NS_; ++k) {
        if (k + 1 < NS_) {
            wait_ds0();   // buffer (k+1)&1 fully consumed (stage k-1) before DMA rewrites it
            stage_load(xicB, dyB, sXic[wave][(k + 1) & 1], sDy[wave][(k + 1) & 1],
                       lane, base + (k + 1) * S_);
            wait_async16();   // in-order completion: stage k's 16 loads have landed
        } else {
            wait_async0();
        }
        const float* xr = sXic[wave][k & 1] + lane * XSTRIDE;
        const float* dr = sDy [wave][k & 1] + lane * DSTRIDE;
#pragma unroll
        for (int r = 0; r < S_; ++r) {
            float x0 = xr[r * 4 + 0], x1 = xr[r * 4 + 1];
            float x2 = xr[r * 4 + 2], x3 = xr[r * 4 + 3];
            float d0 = dr[r * 2 + 0], d1 = dr[r * 2 + 1];
            float G00 = Aeff[0][0] - (x0 * Cd[0][0] + x1 * Cd[1][0]);
            float G01 = Aeff[0][1] - (x0 * Cd[0][1] + x1 * Cd[1][1]);
            float G10 = Aeff[1][0] - (x2 * Cd[0][0] + x3 * Cd[1][0]);
            float G11 = Aeff[1][1] - (x2 * Cd[0][1] + x3 * Cd[1][1]);
            float u0 = x0 * d0 + x1 * d1;
            float u1 = x2 * d0 + x3 * d1;
            float nM00 = G00 * M00 + G01 * M10, nM01 = G00 * M01 + G01 * M11;
            float nM10 = G10 * M00 + G11 * M10, nM11 = G10 * M01 + G11 * M11;
            float nv0  = G00 * v0 + G01 * v1 + u0;
            float nv1  = G10 * v0 + G11 * v1 + u1;
            M00 = nM00; M01 = nM01; M10 = nM10; M11 = nM11; v0 = nv0; v1 = nv1;
        }
    }

    const int c = cg * CPW + lane;
    float* o = chunkMV + ((size_t)b * NCHUNK + c) * 6;
    o[0] = M00; o[1] = M01; o[2] = M10; o[3] = M11; o[4] = v0; o[5] = v1;
}

// ---------------- phase 2: scan chunk affines -> chunk-start states ----------------
__global__ __launch_bounds__(256) void gr_phase2(const float* __restrict__ chunkMV,
                                                 float* __restrict__ xstart) {
    int b = blockIdx.x * blockDim.x + threadIdx.x;
    if (b >= B_) return;
    float x0 = 1.0f, x1 = 0.0f;                 // initial state [1,0]
    const float* p = chunkMV + (size_t)b * NCHUNK * 6;
    float*       o = xstart  + (size_t)b * NCHUNK * 2;
    for (int c = 0; c < NCHUNK; ++c) {
        o[c * 2 + 0] = x0; o[c * 2 + 1] = x1;   // state BEFORE chunk c
        float M00 = p[c * 6 + 0], M01 = p[c * 6 + 1];
        float M10 = p[c * 6 + 2], M11 = p[c * 6 + 3];
        float w0  = p[c * 6 + 4], w1  = p[c * 6 + 5];
        float nx0 = M00 * x0 + M01 * x1 + w0;
        float nx1 = M10 * x0 + M11 * x1 + w1;
        x0 = nx0; x1 = nx1;
    }
}

// ---------------- phase 3: replay chunks with correct start state, emit outputs ----------------
__global__ __launch_bounds__(128) void gr_phase3(const float* __restrict__ xic,
                                                 const float* __restrict__ dy,
                                                 const float* __restrict__ A4,
                                                 const float* __restrict__ C4,
                                                 const float* __restrict__ xstart,
                                                 float* __restrict__ out) {
    __shared__ float sXic[WPB][2][CPW * XSTRIDE];
    __shared__ float sDy [WPB][2][CPW * DSTRIDE];

    const int wave = threadIdx.x >> 5, lane = threadIdx.x & 31;
    const int wgid = blockIdx.x * WPB + wave;
    const int b    = wgid >> 2;
    const int cg   = wgid & 3;
    const int base = cg * CPW * L_;

    const float* xicB = xic + (size_t)b * T_ * 4;
    const float* dyB  = dy  + (size_t)b * T_ * 2;
    float*       outB = out + (size_t)b * T_ * 2;

    float Aeff[2][2], Cd[2][2];
#pragma unroll
    for (int i = 0; i < 2; ++i)
#pragma unroll
        for (int j = 0; j < 2; ++j) {
            Aeff[i][j] = A4[i * 2 + j] * DT_ + (i == j ? 1.0f : 0.0f);
            Cd[i][j]   = C4[i * 2 + j] * DT_;
        }

    const int c = cg * CPW + lane;
    float X0 = xstart[((size_t)b * NCHUNK + c) * 2 + 0];
    float X1 = xstart[((size_t)b * NCHUNK + c) * 2 + 1];

    stage_load(xicB, dyB, sXic[wave][0], sDy[wave][0], lane, base);

    for (int k = 0; k < NS_; ++k) {
        if (k + 1 < NS_) {
            wait_ds0();
            stage_load(xicB, dyB, sXic[wave][(k + 1) & 1], sDy[wave][(k + 1) & 1],
                       lane, base + (k + 1) * S_);
            wait_async16();
        } else {
            wait_async0();
        }
        const float* xr = sXic[wave][k & 1] + lane * XSTRIDE;
        const float* dr = sDy [wave][k & 1] + lane * DSTRIDE;
        const int s0 = k * S_;
#pragma unroll
        for (int r = 0; r < S_; ++r) {
            // emit C @ x_prev * DT (pre-update state)
            float o0 = Cd[0][0] * X0 + Cd[0][1] * X1;
            float o1 = Cd[1][0] * X0 + Cd[1][1] * X1;
            int step = c * L_ + s0 + r;
            *(float2*)(outB + (size_t)step * 2) = make_float2(o0, o1);

            float x0 = xr[r * 4 + 0], x1 = xr[r * 4 + 1];
            float x2 = xr[r * 4 + 2], x3 = xr[r * 4 + 3];
            float d0 = dr[r * 2 + 0], d1 = dr[r * 2 + 1];
            float G00 = Aeff[0][0] - (x0 * Cd[0][0] + x1 * Cd[1][0]);
            float G01 = Aeff[0][1] - (x0 * Cd[0][1] + x1 * Cd[1][1]);
            float G10 = Aeff[1][0] - (x2 * Cd[0][0] + x3 * Cd[1][0]);
            float G11 = Aeff[1][1] - (x2 * Cd[0][1] + x3 * Cd[1][1]);
            float u0 = x0 * d0 + x1 * d1;
            float u1 = x2 * d0 + x3 * d1;
            float nX0 = G00 * X0 + G01 * X1 + u0;
            float nX1 = G10 * X0 + G11 * X1 + u1;
            X0 = nX0; X1 = nX1;
        }
    }
}

// ---------------- host launcher ----------------
extern "C" void kernel_launch(void* const* d_in, const int* in_sizes, int n_in,
                              void* d_out, int out_size, void* d_ws, size_t ws_size,
                              hipStream_t stream) {
    (void)in_sizes; (void)n_in; (void)out_size; (void)ws_size;
    const float* xic = (const float*)d_in[0];   // [B,T,2,2]
    const float* dy  = (const float*)d_in[1];   // [B,T,2]
    const float* A4  = (const float*)d_in[2];   // [2,2]
    const float* C4  = (const float*)d_in[3];   // [2,2]
    float* out = (float*)d_out;                 // [B,T,2]

    float* chunkMV = (float*)d_ws;                               // B*NCHUNK*6 floats (1.5 MB)
    float* xstart  = chunkMV + (size_t)B_ * NCHUNK * 6;          // B*NCHUNK*2 floats (0.5 MB)

    const int waves  = (B_ * NCHUNK) / CPW;      // 2048
    const int blocks = waves / WPB;              // 512

    gr_phase1<<<blocks, WPB * 32, 0, stream>>>(xic, dy, A4, C4, chunkMV);
    gr_phase2<<<(B_ + 255) / 256, 256, 0, stream>>>(chunkMV, xstart);
    gr_phase3<<<blocks, WPB * 32, 0, stream>>>(xic, dy, A4, C4, xstart, out);
}